// RGCNConv_55190329754177
// MI455X (gfx1250) — compile-verified
//
#include <hip/hip_runtime.h>
#include <hip/hip_bf16.h>

#define N_NODES   100000
#define IN_CH     128
#define OUT_CH    128
#define N_REL     8
#define N_BASES   4
#define E_PER_REL 100000

typedef float v2f __attribute__((ext_vector_type(2)));
typedef float v8f __attribute__((ext_vector_type(8)));

// D = A(16x4 f32) * B(4x16 f32) + C(16x16 f32), wave32 WMMA
__device__ __forceinline__ v8f wmma4(v2f a, v2f b, v8f c) {
    return __builtin_amdgcn_wmma_f32_16x16x4_f32(
        /*neg_a=*/false, a, /*neg_b=*/false, b,
        /*c_mod=*/(short)0, c, /*reuse_a=*/false, /*reuse_b=*/false);
}

// ---- W_r = sum_b coeffs[r,b] * bases[b]  (8 x 128 x 128) ----
__global__ void k_combine(const float* __restrict__ bases,
                          const float* __restrict__ coeffs,
                          float* __restrict__ W_all) {
    int tid = blockIdx.x * blockDim.x + threadIdx.x;
    if (tid >= N_REL * IN_CH * OUT_CH) return;
    int r = tid >> 14;
    int j = tid & 16383;
    float s = 0.f;
#pragma unroll
    for (int b = 0; b < N_BASES; ++b)
        s += coeffs[r * N_BASES + b] * bases[b * (IN_CH * OUT_CH) + j];
    W_all[tid] = s;
}

// ---- per-relation in-degree histogram (u32 atomics) ----
__global__ void k_hist(const long long* __restrict__ ei,
                       unsigned* __restrict__ cnt) {
    int tid = blockIdx.x * blockDim.x + threadIdx.x;
    if (tid >= N_REL * E_PER_REL) return;
    int r = tid / E_PER_REL;
    int e = tid - r * E_PER_REL;
    int row = (int)ei[(size_t)(r * 2) * E_PER_REL + e];
    atomicAdd(&cnt[(size_t)r * N_NODES + row], 1u);
}

// ---- counts -> 1/max(deg,1) in place ----
__global__ void k_inv(unsigned* __restrict__ cnt) {
    int tid = blockIdx.x * blockDim.x + threadIdx.x;
    if (tid >= N_REL * N_NODES) return;
    unsigned c = cnt[tid];
    ((float*)cnt)[tid] = 1.0f / (float)(c ? c : 1u);
}

// LDS layout: sWp[(k>>1)*256 + n*2 + (k&1)] = B[k][n]
// => v2f view bp[p*128 + n] = (B[2p][n], B[2p+1][n]), one ds_load_b64.

// ---- messages = x[col] @ W_r ; atomic scatter into out[row] ----
__global__ void __launch_bounds__(256)
k_msg(const float* __restrict__ x,
      const long long* __restrict__ rowIdx,
      const long long* __restrict__ colIdx,
      const float* __restrict__ W,       // [128 k][128 o] row-major
      float* __restrict__ out,
      int nTiles) {
    __shared__ float sWp[IN_CH * OUT_CH];  // 64 KB, pair-interleaved
    for (int i = threadIdx.x * 4; i < IN_CH * OUT_CH; i += 256 * 4) {
        float4 w = *reinterpret_cast<const float4*>(W + i);
        int k = i >> 7, n = i & 127;           // n multiple of 4
        int base = (k >> 1) * 256 + (k & 1);
        sWp[base + (n + 0) * 2] = w.x;
        sWp[base + (n + 1) * 2] = w.y;
        sWp[base + (n + 2) * 2] = w.z;
        sWp[base + (n + 3) * 2] = w.w;
    }
    __syncthreads();

    const int wave = threadIdx.x >> 5;
    const int lane = threadIdx.x & 31;
    const int tile = blockIdx.x * 8 + wave;
    if (tile >= nTiles) return;

    const int m  = lane & 15;   // tile-local edge idx / col within col-tile
    const int hi = lane >> 4;   // lane half

    const int e   = tile * 16 + m;
    const int row = (int)rowIdx[e];
    const int col = (int)colIdx[e];
    const float* xrow = x + (size_t)col * IN_CH + 4 * hi;
    const v2f* bp = reinterpret_cast<const v2f*>(sWp);

    v8f acc[8] = {};
    for (int s8 = 0; s8 < IN_CH; s8 += 8) {
        // lane-half loads its 4 K values for this 8-K block in one float4
        float4 av = *reinterpret_cast<const float4*>(xrow + s8);
        v2f a0 = {av.x, av.y};
        v2f a1 = {av.z, av.w};
        const int p0 = (s8 >> 1) + 2 * hi;     // K-pair index, step A
#pragma unroll
        for (int ct = 0; ct < 8; ++ct)
            acc[ct] = wmma4(a0, bp[p0 * 128 + ct * 16 + m], acc[ct]);
#pragma unroll
        for (int ct = 0; ct < 8; ++ct)
            acc[ct] = wmma4(a1, bp[(p0 + 1) * 128 + ct * 16 + m], acc[ct]);
    }

#pragma unroll
    for (int j = 0; j < 8; ++j) {
        int mm = j + 8 * hi;                 // output-tile row 0..15
        int r2 = __shfl(row, mm, 32);        // dst node of that edge
        float* orow = out + (size_t)r2 * OUT_CH;
#pragma unroll
        for (int ct = 0; ct < 8; ++ct) {
            __hip_atomic_fetch_add(&orow[ct * 16 + m], acc[ct][j],
                                   __ATOMIC_RELAXED, __HIP_MEMORY_SCOPE_AGENT);
        }
    }
}

// ---- out *= 1/deg (per relation, cumulative quirk of reference) ----
__global__ void k_scale(float* __restrict__ out, const float* __restrict__ dinv) {
    int tid = blockIdx.x * blockDim.x + threadIdx.x;
    if (tid >= N_NODES * OUT_CH) return;
    out[tid] *= dinv[tid >> 7];
}

// ---- out += x @ selfW^T + bias ----
__global__ void __launch_bounds__(256)
k_self(const float* __restrict__ x,
       const float* __restrict__ selfW,     // [OUT][IN]; B[k][o] = selfW[o][k]
       const float* __restrict__ bias,
       float* __restrict__ out,
       int nTiles) {
    __shared__ float sWp[IN_CH * OUT_CH];   // pair-interleaved transposed
    for (int i = threadIdx.x * 4; i < IN_CH * OUT_CH; i += 256 * 4) {
        float4 w = *reinterpret_cast<const float4*>(selfW + i);
        int o = i >> 7, k = i & 127;         // k multiple of 4
        sWp[((k >> 1) + 0) * 256 + o * 2 + 0] = w.x;
        sWp[((k >> 1) + 0) * 256 + o * 2 + 1] = w.y;
        sWp[((k >> 1) + 1) * 256 + o * 2 + 0] = w.z;
        sWp[((k >> 1) + 1) * 256 + o * 2 + 1] = w.w;
    }
    __syncthreads();

    const int wave = threadIdx.x >> 5;
    const int lane = threadIdx.x & 31;
    const int tile = blockIdx.x * 8 + wave;
    if (tile >= nTiles) return;

    const int m  = lane & 15;
    const int hi = lane >> 4;
    const int node = tile * 16 + m;
    const float* xrow = x + (size_t)node * IN_CH + 4 * hi;
    const v2f* bp = reinterpret_cast<const v2f*>(sWp);

    v8f acc[8] = {};
    for (int s8 = 0; s8 < IN_CH; s8 += 8) {
        float4 av = *reinterpret_cast<const float4*>(xrow + s8);
        v2f a0 = {av.x, av.y};
        v2f a1 = {av.z, av.w};
        const int p0 = (s8 >> 1) + 2 * hi;
#pragma unroll
        for (int ct = 0; ct < 8; ++ct)
            acc[ct] = wmma4(a0, bp[p0 * 128 + ct * 16 + m], acc[ct]);
#pragma unroll
        for (int ct = 0; ct < 8; ++ct)
            acc[ct] = wmma4(a1, bp[(p0 + 1) * 128 + ct * 16 + m], acc[ct]);
    }

#pragma unroll
    for (int ct = 0; ct < 8; ++ct) {
        float bv = bias[ct * 16 + m];
#pragma unroll
        for (int j = 0; j < 8; ++j) {
            int mm = j + 8 * hi;
            size_t idx = (size_t)(tile * 16 + mm) * OUT_CH + ct * 16 + m;
            out[idx] += acc[ct][j] + bv;
        }
    }
}

extern "C" void kernel_launch(void* const* d_in, const int* in_sizes, int n_in,
                              void* d_out, int out_size, void* d_ws, size_t ws_size,
                              hipStream_t stream) {
    (void)in_sizes; (void)n_in; (void)out_size; (void)ws_size;
    const float*     x      = (const float*)d_in[0];
    const long long* ei     = (const long long*)d_in[1];   // int64 [R,2,E]
    const float*     bases  = (const float*)d_in[2];
    const float*     coeffs = (const float*)d_in[3];
    const float*     selfW  = (const float*)d_in[4];
    const float*     bias   = (const float*)d_in[5];
    float*           out    = (float*)d_out;

    char*  ws    = (char*)d_ws;
    float* W_all = (float*)ws;                                           // 512 KB
    float* deg   = (float*)(ws + (size_t)N_REL * IN_CH * OUT_CH * 4);    // 3.05 MB

    hipMemsetAsync(out, 0, (size_t)N_NODES * OUT_CH * sizeof(float), stream);
    hipMemsetAsync(deg, 0, (size_t)N_REL * N_NODES * sizeof(float), stream);

    k_combine<<<(N_REL * IN_CH * OUT_CH + 255) / 256, 256, 0, stream>>>(bases, coeffs, W_all);
    k_hist<<<(N_REL * E_PER_REL + 255) / 256, 256, 0, stream>>>(ei, (unsigned*)deg);
    k_inv<<<(N_REL * N_NODES + 255) / 256, 256, 0, stream>>>((unsigned*)deg);

    const int nTilesE = E_PER_REL / 16;        // 6250, exact
    const int nTilesN = N_NODES / 16;          // 6250, exact
    for (int r = 0; r < N_REL; ++r) {
        k_msg<<<(nTilesE + 7) / 8, 256, 0, stream>>>(
            x,
            ei + (size_t)(r * 2 + 0) * E_PER_REL,   // row (dst)
            ei + (size_t)(r * 2 + 1) * E_PER_REL,   // col (src)
            W_all + (size_t)r * IN_CH * OUT_CH,
            out, nTilesE);
        k_scale<<<(N_NODES * OUT_CH + 255) / 256, 256, 0, stream>>>(
            out, deg + (size_t)r * N_NODES);
    }
    k_self<<<(nTilesN + 7) / 8, 256, 0, stream>>>(x, selfW, bias, out, nTilesN);
}